// Encoder_12017318494790
// MI455X (gfx1250) — compile-verified
//
#include <hip/hip_runtime.h>

typedef __attribute__((ext_vector_type(2))) float v2f;
typedef __attribute__((ext_vector_type(8))) float v8f;

#define D 128

// ---------------- degree / normalization ----------------
__global__ void k_deg_init(float* __restrict__ deg, int n) {
  int i = blockIdx.x * blockDim.x + threadIdx.x;
  if (i < n) deg[i] = 1.0f;  // self-loop contributes 1 to every node's degree
}

__global__ void k_deg_accum(const long long* __restrict__ dst,
                            float* __restrict__ deg, int E) {
  int e = blockIdx.x * blockDim.x + threadIdx.x;
  if (e < E) unsafeAtomicAdd(&deg[(int)dst[e]], 1.0f);
}

__global__ void k_dinv(float* __restrict__ deg, int n) {
  int i = blockIdx.x * blockDim.x + threadIdx.x;
  if (i < n) {
    float d = deg[i];
    deg[i] = (d > 0.0f) ? rsqrtf(d) : 0.0f;
  }
}

// ---------------- fp32 WMMA GEMM: Y[n x 128] = X[n x 128] @ W[128 x 128] (+bias) ----------------
// block = 256 threads (8 waves). Each wave computes 16 rows x 128 cols.
// W staged in LDS swizzled as k-pairs: lW2[(k/2)*128 + n] = {W[k][n], W[k+1][n]}
// so each B fragment is a single aligned ds_load_b64 into an even VGPR pair.
__global__ __launch_bounds__(256) void k_gemm(const float* __restrict__ X,
                                              const float* __restrict__ W,
                                              const float* __restrict__ bias,
                                              float* __restrict__ Y, int n) {
  __shared__ __align__(16) v2f lW2[(D / 2) * D];  // 64 KB of the 320 KB/WGP LDS
  for (int idx = threadIdx.x; idx < (D / 2) * D; idx += 256) {
    const int p   = idx >> 7;    // k-pair index 0..63
    const int col = idx & 127;
    v2f v;
    v.x = W[(2 * p) * D + col];
    v.y = W[(2 * p + 1) * D + col];
    lW2[idx] = v;
  }
  __syncthreads();

  const int wave = threadIdx.x >> 5;
  const int lane = threadIdx.x & 31;
  const int half = lane >> 4;   // 0: lanes 0-15, 1: lanes 16-31
  const int l16  = lane & 15;
  const int row0 = blockIdx.x * 128 + wave * 16;

  v8f acc[8] = {};

  // A-frag row for this lane (clamped; garbage rows are never stored)
  int arow = row0 + l16;
  if (arow >= n) arow = n - 1;
  const float* aptr = X + (size_t)arow * D + 2 * half;  // reg0=A[m][k+2h], reg1=A[m][k+2h+1]

  for (int k = 0; k < D; k += 4) {
    const v2f av = *(const v2f*)(aptr + k);             // one global_load_b64
    const int prow = (k >> 1) + half;                   // (k + 2*half)/2
    const v2f* brow = lW2 + prow * D + l16;
#pragma unroll
    for (int t = 0; t < 8; ++t) {
      const v2f bv = brow[t * 16];                      // one ds_load_b64
      acc[t] = __builtin_amdgcn_wmma_f32_16x16x4_f32(
          /*neg_a=*/false, av, /*neg_b=*/false, bv,
          /*c_mod=*/(short)0, acc[t], /*reuse_a=*/false, /*reuse_b=*/false);
    }
  }

  // D layout: VGPR r -> M = r + 8*half, N = l16
#pragma unroll
  for (int t = 0; t < 8; ++t) {
    const int col = t * 16 + l16;
    const float badd = bias ? bias[col] : 0.0f;
#pragma unroll
    for (int r = 0; r < 8; ++r) {
      const int row = row0 + r + 8 * half;
      if (row < n) Y[(size_t)row * D + col] = acc[t][r] + badd;
    }
  }
}

// ---------------- aggregation ----------------
// a[i,:] = h[i,:] * dinv[i]^2   (self-loop term, atomic-free init)
__global__ void k_selfinit(const float* __restrict__ h, const float* __restrict__ dinv,
                           float* __restrict__ a, long long total) {
  long long i = (long long)blockIdx.x * blockDim.x + threadIdx.x;
  if (i < total) {
    float w = dinv[(int)(i >> 7)];
    a[i] = h[i] * (w * w);
  }
}

// one wave (32 lanes) per edge, 4 floats per lane:
//   a[dst,:] += h[src,:] * dinv[src]*dinv[dst]
// edge id is wave-uniform -> readfirstlane so metadata comes in via scalar loads.
__global__ __launch_bounds__(256) void k_scatter(const long long* __restrict__ src,
                                                 const long long* __restrict__ dst,
                                                 const float* __restrict__ dinv,
                                                 const float* __restrict__ h,
                                                 float* __restrict__ a, int E) {
  int e = blockIdx.x * 8 + (threadIdx.x >> 5);
  if (e >= E) return;                      // wave-uniform branch
  e = __builtin_amdgcn_readfirstlane(e);   // force SGPR -> scalar loads below
  const int lane = threadIdx.x & 31;
  const int s = (int)src[e];
  const int d = (int)dst[e];
  const float w = dinv[s] * dinv[d];
  const int f = lane * 4;
  const float4 hv = *(const float4*)(h + (size_t)s * D + f);
  float* ap = a + (size_t)d * D + f;
  unsafeAtomicAdd(ap + 0, hv.x * w);
  unsafeAtomicAdd(ap + 1, hv.y * w);
  unsafeAtomicAdd(ap + 2, hv.z * w);
  unsafeAtomicAdd(ap + 3, hv.w * w);
}

// o = relu(a + b)
__global__ void k_bias_relu(const float* __restrict__ a, const float* __restrict__ b,
                            float* __restrict__ o, long long total) {
  long long i = (long long)blockIdx.x * blockDim.x + threadIdx.x;
  if (i < total) {
    float v = a[i] + b[(int)(i & (D - 1))];
    o[i] = v > 0.0f ? v : 0.0f;
  }
}

// out += relu(a + b2)
__global__ void k_final(const float* __restrict__ a, const float* __restrict__ b2,
                        float* __restrict__ out, long long total) {
  long long i = (long long)blockIdx.x * blockDim.x + threadIdx.x;
  if (i < total) {
    float v = a[i] + b2[(int)(i & (D - 1))];
    v = v > 0.0f ? v : 0.0f;
    out[i] += v;
  }
}

extern "C" void kernel_launch(void* const* d_in, const int* in_sizes, int n_in,
                              void* d_out, int out_size, void* d_ws, size_t ws_size,
                              hipStream_t stream) {
  const float*     x   = (const float*)d_in[0];
  const long long* ei  = (const long long*)d_in[1];
  const float*     W1  = (const float*)d_in[2];
  const float*     b1  = (const float*)d_in[3];
  const float*     W2  = (const float*)d_in[4];
  const float*     b2  = (const float*)d_in[5];
  const float*     Wfc = (const float*)d_in[6];
  const float*     bfc = (const float*)d_in[7];

  const int n = in_sizes[0] / D;
  const int E = in_sizes[1] / 2;
  const long long* srcp = ei;
  const long long* dstp = ei + E;

  float* dinv = (float*)d_ws;
  float* B    = dinv + (((size_t)n + 255) & ~(size_t)255);
  float* C    = B + (size_t)n * D;
  float* out  = (float*)d_out;

  const long long total = (long long)n * D;
  const int nGrid  = (n + 255) / 256;
  const int eGrid  = (E + 255) / 256;
  const int elGrid = (int)((total + 255) / 256);
  const int scGrid = (E + 7) / 8;          // 8 edges (waves) per 256-thread block
  const int gmGrid = (n + 127) / 128;

  // normalization: deg (with self-loop) -> deg^{-1/2}
  k_deg_init<<<nGrid, 256, 0, stream>>>(dinv, n);
  k_deg_accum<<<eGrid, 256, 0, stream>>>(dstp, dinv, E);
  k_dinv<<<nGrid, 256, 0, stream>>>(dinv, n);

  // dense projections (fp32 WMMA)
  k_gemm<<<gmGrid, 256, 0, stream>>>(x, W1, nullptr, B, n);     // B = x @ W1
  k_gemm<<<gmGrid, 256, 0, stream>>>(x, Wfc, bfc, out, n);      // out = x @ Wfc + bfc (residual)

  // layer 1 aggregation
  k_selfinit<<<elGrid, 256, 0, stream>>>(B, dinv, C, total);    // C = B * dinv^2
  k_scatter<<<scGrid, 256, 0, stream>>>(srcp, dstp, dinv, B, C, E);
  k_bias_relu<<<elGrid, 256, 0, stream>>>(C, b1, B, total);     // B = relu(C + b1)

  // layer 2
  k_gemm<<<gmGrid, 256, 0, stream>>>(B, W2, nullptr, C, n);     // C = B @ W2
  k_selfinit<<<elGrid, 256, 0, stream>>>(C, dinv, B, total);    // B = C * dinv^2
  k_scatter<<<scGrid, 256, 0, stream>>>(srcp, dstp, dinv, C, B, E);
  k_final<<<elGrid, 256, 0, stream>>>(B, b2, out, total);       // out += relu(B + b2)
}